// GSNN_81896436400579
// MI455X (gfx1250) — compile-verified
//
// GSNN on MI455X (gfx1250): fused per-node-pair layer kernel using
// V_WMMA_F32_16X16X4_F32 for both sparse-dense contractions.
//
// Latency-bound problem (~0.35 GFLOP, ~25MB working set, all L2-resident at
// 192MB). x_edge stored [E][128] f32 (edge-major, 512B coalesced rows),
// double-buffered across layers (4 grid launches = global barriers).
// One workgroup (128 thr = 4 waves) per node PAIR; each wave handles 2 batch
// tiles of 16. Masked theta columns are pre-expanded into LDS so the WMMA hot
// loops are branch-free (pure ds_load + global_load + v_wmma); CSR lists are
// padded to the WMMA K/M granularity with zero-theta records; all hot-loop
// addresses are 32-bit (saddr + voffset form). Each func-src edge has a
// unique writer -> no atomics in the hot path.

#include <hip/hip_runtime.h>
#include <math.h>

typedef float v2f __attribute__((ext_vector_type(2)));
typedef float v8f __attribute__((ext_vector_type(8)));

#define N_FUNC 2000
#define N_IN   500
#define N_OUT  500
#define NPAIR  (N_FUNC / 2)
#define CH     8
#define BSZ    128
#define LAYERS 4
#define LN_EPS 1e-5f
#define IN_CAP  128   // max (padded) in-edges per node pair staged in LDS
#define OUT_CAP 128   // max (padded) out-edges per node pair staged in LDS

// ---------------- setup kernels (per-call structure build) -------------------

__global__ void k_init_maps(int* t1ofs, int* t3ofs, int* cntIn, int* cntOut, int E) {
  int i = blockIdx.x * 256 + threadIdx.x;
  if (i < E) { t1ofs[i] = -1; t3ofs[i] = -1; }
  if (i < NPAIR) { cntIn[i] = 0; cntOut[i] = 0; }
}

// w1/w3 entries come in groups of 8 (one edge x 8 channels): record the theta
// base offset per edge.
__global__ void k_build_t1(const int* __restrict__ w1_src, int* t1ofs, int G) {
  int j = blockIdx.x * 256 + threadIdx.x;
  if (j < G) t1ofs[w1_src[8 * j]] = 8 * j;
}
__global__ void k_build_t3(const int* __restrict__ w3_dst, int* t3ofs, int G) {
  int j = blockIdx.x * 256 + threadIdx.x;
  if (j < G) t3ofs[w3_dst[8 * j]] = 8 * j;
}

__global__ void k_count(const int* __restrict__ edge_src, const int* __restrict__ edge_dst,
                        const int* __restrict__ t1ofs, const int* __restrict__ t3ofs,
                        int* cntIn, int* cntOut, int E) {
  int e = blockIdx.x * 256 + threadIdx.x;
  if (e >= E) return;
  if (t1ofs[e] >= 0) atomicAdd(&cntIn[edge_dst[e] >> 1], 1);
  if (t3ofs[e] >= 0) atomicAdd(&cntOut[edge_src[e] >> 1], 1);
}

// single-block Hillis-Steele scan over NPAIR(<=1024) PADDED counts, twice.
// In-lists padded to multiples of 4 (WMMA K step), out-lists to 16 (M tile).
__global__ void k_scan(const int* __restrict__ cntIn, const int* __restrict__ cntOut,
                       int* ofsIn, int* ofsOut, int* curIn, int* curOut) {
  __shared__ int buf[2][1024];
  int t = threadIdx.x;
  // --- in-edge counts (pad to x4) ---
  int cp = (t < NPAIR) ? ((cntIn[t] + 3) & ~3) : 0;
  buf[0][t] = cp;
  __syncthreads();
  int s = 0;
  for (int off = 1; off < 1024; off <<= 1) {
    int v = buf[s][t] + ((t >= off) ? buf[s][t - off] : 0);
    buf[1 ^ s][t] = v;
    s ^= 1;
    __syncthreads();
  }
  if (t < NPAIR) {
    int excl = buf[s][t] - cp;
    ofsIn[t] = excl; curIn[t] = excl;
    if (t == NPAIR - 1) ofsIn[NPAIR] = buf[s][t];
  }
  __syncthreads();
  // --- out-edge counts (pad to x16) ---
  cp = (t < NPAIR) ? ((cntOut[t] + 15) & ~15) : 0;
  buf[0][t] = cp;
  __syncthreads();
  s = 0;
  for (int off = 1; off < 1024; off <<= 1) {
    int v = buf[s][t] + ((t >= off) ? buf[s][t - off] : 0);
    buf[1 ^ s][t] = v;
    s ^= 1;
    __syncthreads();
  }
  if (t < NPAIR) {
    int excl = buf[s][t] - cp;
    ofsOut[t] = excl; curOut[t] = excl;
    if (t == NPAIR - 1) ofsOut[NPAIR] = buf[s][t];
  }
}

// pre-fill record arrays with pad records (overwritten by k_fill where real):
// in pad  = {edge 0, tofs -1}  -> staged A column all zeros, harmless B addr
// out pad = {-1, 0}            -> staged A row all zeros, store skipped
__global__ void k_pad(int2* inRec, int nInAlloc, int2* outRec, int nOutAlloc) {
  int i = blockIdx.x * 256 + threadIdx.x;
  if (i < nInAlloc)  inRec[i]  = make_int2(0, -1);
  if (i < nOutAlloc) outRec[i] = make_int2(-1, 0);
}

// record = { edge | (owner_in_pair << 20), theta_base_offset }
__global__ void k_fill(const int* __restrict__ edge_src, const int* __restrict__ edge_dst,
                       const int* __restrict__ t1ofs, const int* __restrict__ t3ofs,
                       int* curIn, int* curOut, int2* inRec, int2* outRec, int E) {
  int e = blockIdx.x * 256 + threadIdx.x;
  if (e >= E) return;
  int t1 = t1ofs[e];
  if (t1 >= 0) {
    int d = edge_dst[e];
    int slot = atomicAdd(&curIn[d >> 1], 1);
    inRec[slot] = make_int2(e | ((d & 1) << 20), t1);
  }
  int t3 = t3ofs[e];
  if (t3 >= 0) {
    int sc = edge_src[e];
    int slot = atomicAdd(&curOut[sc >> 1], 1);
    outRec[slot] = make_int2(e | ((sc & 1) << 20), t3);
  }
}

// x_edge init: only input-src edges nonzero; write BOTH buffers (input edges
// are never written by the layer kernel -> clamped to x0 forever).
__global__ void k_init_x(const float* __restrict__ x, const int* __restrict__ edge_src,
                         float* xeA, float* xeB) {
  int e = blockIdx.x;
  int b = threadIdx.x;
  int s = edge_src[e];
  float v = 0.f;
  if (s >= N_FUNC && s < N_FUNC + N_IN) v = x[b * N_IN + (s - N_FUNC)];
  xeA[(size_t)e * BSZ + b] = v;
  xeB[(size_t)e * BSZ + b] = v;
}

// ---------------- fused layer kernel (WMMA) ---------------------------------

__global__ __launch_bounds__(128) void k_layer(
    const float* __restrict__ xin, float* __restrict__ xout,
    const float* __restrict__ theta1, const float* __restrict__ bias1,
    const float* __restrict__ gma, const float* __restrict__ bta,
    const float* __restrict__ theta3, const float* __restrict__ bias3,
    const int2* __restrict__ inRec, const int* __restrict__ ofsIn,
    const int2* __restrict__ outRec, const int* __restrict__ ofsOut) {
  __shared__ float sInA[IN_CAP][16];   // A[m][k] columns for lin1 (theta1, masked)
  __shared__ int   sInO[IN_CAP];       // x-row element offset (e*BSZ) per in-slot
  __shared__ float sOutA[OUT_CAP][16]; // A[m][k] rows for lin3 (theta3, masked)
  __shared__ int   sOutO[OUT_CAP];     // x-row element offset per out-slot (-1 = pad)
  __shared__ float sOutB[OUT_CAP];     // bias3 per out-slot

  int p = blockIdx.x;
  int tid = threadIdx.x;
  int inBase  = ofsIn[p];
  int nIn     = min(ofsIn[p + 1] - inBase, IN_CAP);    // padded multiple of 4
  int outBase = ofsOut[p];
  int nOut    = min(ofsOut[p + 1] - outBase, OUT_CAP); // padded multiple of 16

  // ---- stage masked theta operands into LDS (once per block) --------------
  for (int i = tid; i < nIn; i += 128) {
    int2 r = inRec[inBase + i];
    sInO[i] = (r.x & 0xFFFFF) * BSZ;   // pre-scaled 32-bit row offset
    int own = (r.x >> 20) & 1;
    #pragma unroll
    for (int c = 0; c < 16; ++c) sInA[i][c] = 0.f;
    if (r.y >= 0) {
      #pragma unroll
      for (int c = 0; c < 8; ++c) sInA[i][own * 8 + c] = theta1[r.y + c];
    }
  }
  for (int i = tid; i < nOut; i += 128) {
    int2 r = outRec[outBase + i];
    #pragma unroll
    for (int c = 0; c < 16; ++c) sOutA[i][c] = 0.f;
    if (r.x >= 0) {
      int e = r.x & 0xFFFFF;
      int own = (r.x >> 20) & 1;
      sOutO[i] = e * BSZ;              // pre-scaled 32-bit row offset
      sOutB[i] = bias3[e];
      #pragma unroll
      for (int c = 0; c < 8; ++c) sOutA[i][own * 8 + c] = theta3[r.y + c];
    } else {
      sOutO[i] = -1;
      sOutB[i] = 0.f;
    }
  }
  __syncthreads();

  int lane = tid & 31;
  int wave = tid >> 5;
  int m = lane & 15;          // A row M / B column N (batch-in-tile)
  bool up = lane >= 16;
  int ksel = up ? 2 : 0;      // A/B layout: v0 holds K={0|2}, v1 holds K={1|3}
  int node = 2 * p + (up ? 1 : 0);

  float b1[8], g[8], bb[8];
  #pragma unroll
  for (int r = 0; r < 8; ++r) {
    b1[r] = bias1[node * 8 + r];
    g[r]  = gma[node * 8 + r];
    bb[r] = bta[node * 8 + r];
  }

  int nChunk = nIn >> 2;
  int nTile  = nOut >> 4;

  for (int bt = wave; bt < BSZ / 16; bt += 4) {   // 2 batch tiles per wave
    unsigned bm = (unsigned)(bt * 16 + m);        // batch element of this lane

    // ---- lin1: D[16ch x 16b] += Theta1^T[16 x 4] x X[4 x 16b], branch-free -
    v8f acc = {0.f, 0.f, 0.f, 0.f, 0.f, 0.f, 0.f, 0.f};
    for (int c = 0; c < nChunk; ++c) {
      int kb = c * 4 + ksel;
      v2f a = {sInA[kb][m], sInA[kb + 1][m]};
      unsigned o0 = (unsigned)sInO[kb] + bm;      // 32-bit addressing
      unsigned o1 = (unsigned)sInO[kb + 1] + bm;
      v2f b = {xin[o0], xin[o1]};
      acc = __builtin_amdgcn_wmma_f32_16x16x4_f32(false, a, false, b,
                                                  (short)0, acc, false, false);
    }

    // ---- bias1 + LayerNorm(8ch) + exact GELU, all per-lane ----------------
    float h[8];
    float mu = 0.f;
    #pragma unroll
    for (int r = 0; r < 8; ++r) { h[r] = acc[r] + b1[r]; mu += h[r]; }
    mu *= 0.125f;
    float var = 0.f;
    #pragma unroll
    for (int r = 0; r < 8; ++r) { float d = h[r] - mu; var += d * d; }
    var *= 0.125f;
    float inv = rsqrtf(var + LN_EPS);
    #pragma unroll
    for (int r = 0; r < 8; ++r) {
      float t = (h[r] - mu) * inv * g[r] + bb[r];
      h[r] = 0.5f * t * (1.f + erff(t * 0.70710678118654752f));
    }

    // ---- lin3 B operand H[16ch x 16b]: half-wave swaps --------------------
    float hs[8];
    #pragma unroll
    for (int r = 0; r < 8; ++r) hs[r] = __shfl_xor(h[r], 16);
    v2f Bc[4];
    Bc[0].x = up ? hs[2] : h[0];  Bc[0].y = up ? hs[3] : h[1];   // K 0..3  (d1)
    Bc[1].x = up ? hs[6] : h[4];  Bc[1].y = up ? hs[7] : h[5];   // K 4..7  (d1)
    Bc[2].x = up ? h[2] : hs[0];  Bc[2].y = up ? h[3] : hs[1];   // K 8..11 (d2)
    Bc[3].x = up ? h[6] : hs[4];  Bc[3].y = up ? h[7] : hs[5];   // K 12..15(d2)

    // ---- lin3: D[16e x 16b] = Theta3[16e x 16] x H[16 x 16b] --------------
    for (int tl = 0; tl < nTile; ++tl) {
      int ri = tl * 16 + m;
      v8f d = {0.f, 0.f, 0.f, 0.f, 0.f, 0.f, 0.f, 0.f};
      #pragma unroll
      for (int t = 0; t < 4; ++t) {
        v2f a = *(const v2f*)&sOutA[ri][t * 4 + ksel];  // 8B-aligned ds_load_b64
        d = __builtin_amdgcn_wmma_f32_16x16x4_f32(false, a, false, Bc[t],
                                                  (short)0, d, false, false);
      }
      // epilogue: residual + bias3; unique writer per edge (no atomics).
      int rbase = (up ? 8 : 0) + tl * 16;
      #pragma unroll
      for (int r = 0; r < 8; ++r) {
        int eo = sOutO[rbase + r];
        if (eo >= 0) {
          float bi = sOutB[rbase + r];
          unsigned idx = (unsigned)eo + bm;       // 32-bit addressing
          xout[idx] = xin[idx] + bi + d[r];
        }
      }
    }
  }
}

// ---------------- readout ---------------------------------------------------

__global__ void k_zero_out(float* out, int n) {
  int i = blockIdx.x * 256 + threadIdx.x;
  if (i < n) out[i] = 0.f;
}

__global__ void k_readout(const float* __restrict__ xe, const int* __restrict__ edge_dst,
                          float* out) {
  int e = blockIdx.x;
  int b = threadIdx.x;
  int d = edge_dst[e];
  if (d >= N_FUNC + N_IN)
    atomicAdd(&out[b * N_OUT + (d - (N_FUNC + N_IN))],
              xe[(size_t)e * BSZ + b] * (1.f / LAYERS));
}

// ---------------- host launcher ---------------------------------------------

extern "C" void kernel_launch(void* const* d_in, const int* in_sizes, int n_in,
                              void* d_out, int out_size, void* d_ws, size_t ws_size,
                              hipStream_t stream) {
  const float* x      = (const float*)d_in[0];
  const float* theta1 = (const float*)d_in[1];
  const float* bias1  = (const float*)d_in[2];
  const float* gma    = (const float*)d_in[3];
  const float* bta    = (const float*)d_in[4];
  const float* theta3 = (const float*)d_in[5];
  const float* bias3  = (const float*)d_in[6];
  const int* edge_src = (const int*)d_in[7];
  const int* edge_dst = (const int*)d_in[8];
  const int* w1_src   = (const int*)d_in[9];
  const int* w3_dst   = (const int*)d_in[12];

  const int E  = in_sizes[7];
  const int K1 = in_sizes[9];
  const int K3 = in_sizes[11];
  const int G1 = K1 / 8;
  const int G3 = K3 / 8;
  const int inAlloc  = G1 + 4 * NPAIR;    // worst-case padded totals
  const int outAlloc = G3 + 16 * NPAIR;

  // workspace layout
  char* w = (char*)d_ws;
  size_t off = 0;
  auto alloc = [&](size_t bytes) -> void* {
    void* p = w + off;
    off = (off + bytes + 255) & ~(size_t)255;
    return p;
  };
  float* xeA   = (float*)alloc((size_t)E * BSZ * sizeof(float));
  float* xeB   = (float*)alloc((size_t)E * BSZ * sizeof(float));
  int*   t1ofs = (int*)alloc((size_t)E * sizeof(int));
  int*   t3ofs = (int*)alloc((size_t)E * sizeof(int));
  int*   cntIn = (int*)alloc((size_t)NPAIR * sizeof(int));
  int*   cntOut= (int*)alloc((size_t)NPAIR * sizeof(int));
  int*   ofsIn = (int*)alloc((size_t)(NPAIR + 1) * sizeof(int));
  int*   ofsOut= (int*)alloc((size_t)(NPAIR + 1) * sizeof(int));
  int*   curIn = (int*)alloc((size_t)NPAIR * sizeof(int));
  int*   curOut= (int*)alloc((size_t)NPAIR * sizeof(int));
  int2*  inRec = (int2*)alloc((size_t)inAlloc * sizeof(int2));
  int2*  outRec= (int2*)alloc((size_t)outAlloc * sizeof(int2));
  (void)ws_size;

  int gE = (E + 255) / 256;
  k_init_maps<<<gE, 256, 0, stream>>>(t1ofs, t3ofs, cntIn, cntOut, E);
  k_build_t1<<<(G1 + 255) / 256, 256, 0, stream>>>(w1_src, t1ofs, G1);
  k_build_t3<<<(G3 + 255) / 256, 256, 0, stream>>>(w3_dst, t3ofs, G3);
  k_count<<<gE, 256, 0, stream>>>(edge_src, edge_dst, t1ofs, t3ofs, cntIn, cntOut, E);
  k_scan<<<1, 1024, 0, stream>>>(cntIn, cntOut, ofsIn, ofsOut, curIn, curOut);
  k_pad<<<(max(inAlloc, outAlloc) + 255) / 256, 256, 0, stream>>>(inRec, inAlloc,
                                                                  outRec, outAlloc);
  k_fill<<<gE, 256, 0, stream>>>(edge_src, edge_dst, t1ofs, t3ofs,
                                 curIn, curOut, inRec, outRec, E);
  k_init_x<<<E, BSZ, 0, stream>>>(x, edge_src, xeA, xeB);

  const float* xi = xeA;
  float* xo = xeB;
  for (int l = 0; l < LAYERS; ++l) {
    k_layer<<<NPAIR, 128, 0, stream>>>(xi, xo, theta1, bias1, gma, bta,
                                       theta3, bias3, inRec, ofsIn, outRec, ofsOut);
    const float* t = xi; xi = xo; xo = (float*)t;
  }
  // LAYERS even -> final state back in xeA (== xi here)

  k_zero_out<<<(out_size + 255) / 256, 256, 0, stream>>>((float*)d_out, out_size);
  k_readout<<<E, BSZ, 0, stream>>>(xi, edge_dst, (float*)d_out);
}